// Decoder_89429809037892
// MI455X (gfx1250) — compile-verified
//
#include <hip/hip_runtime.h>

typedef float fx2 __attribute__((ext_vector_type(2)));
typedef float fx8 __attribute__((ext_vector_type(8)));

// ---------------------------------------------------------------------------
// Degree / normalization kernels
// ---------------------------------------------------------------------------
__global__ void deg_init_kernel(float* __restrict__ deg, int n) {
  int i = blockIdx.x * blockDim.x + threadIdx.x;
  if (i < n) deg[i] = 1.0f;  // self-loop weight contributes 1.0 to every node
}

__global__ void deg_accum_kernel(const int* __restrict__ dst,
                                 const float* __restrict__ ew,
                                 float* __restrict__ deg, int e) {
  int i = blockIdx.x * blockDim.x + threadIdx.x;
  if (i < e) unsafeAtomicAdd(&deg[dst[i]], ew[i]);  // native global_atomic_add_f32
}

__global__ void dinv_kernel(const float* __restrict__ deg,
                            float* __restrict__ dinv, int n) {
  int i = blockIdx.x * blockDim.x + threadIdx.x;
  if (i < n) {
    float d = deg[i];
    dinv[i] = (d > 0.0f) ? rsqrtf(d) : 0.0f;
  }
}

// ---------------------------------------------------------------------------
// WMMA GEMM: hs = dinv .* (X @ W); also writes acc = hs (self-loop init).
// One wave computes one 16x16 output tile using V_WMMA_F32_16X16X4_F32
// (full fp32 matrix pipe -- GEMM is only ~1.6 GFLOP total, accuracy is free).
//
// A 16x4 layout (per ISA 7.12.2): lane l holds row m=l%16, K-pair starting at
// kb + 2*(l/16). C/D layout: VGPR r holds M=r+8*(l/16), N=l%16.
// N rows (50000) is an exact multiple of 16, so EXEC is all-ones in every
// active wave (WMMA requirement); the row guard is wave-uniform.
// ---------------------------------------------------------------------------
template <int K, int NOUT>
__global__ void gemm_wmma_scale_kernel(const float* __restrict__ X,
                                       const float* __restrict__ W,
                                       const float* __restrict__ dinv,
                                       float* __restrict__ hs,
                                       float* __restrict__ acc,
                                       int nrows) {
  const int lane = threadIdx.x & 31;
  const int wave = blockIdx.x * (blockDim.x >> 5) + (threadIdx.x >> 5);
  const int ntc  = NOUT / 16;           // tile columns
  const int row0 = (wave / ntc) * 16;
  const int col0 = (wave % ntc) * 16;
  if (row0 >= nrows) return;            // wave-uniform

  const int half = lane >> 4;           // 0: lanes 0-15, 1: lanes 16-31
  const int m    = lane & 15;

  fx8 c = {};
  const float* xrow = X + (size_t)(row0 + m) * K;

#pragma unroll
  for (int kb = 0; kb < K; kb += 4) {
    const int ka = kb + 2 * half;
    fx2 a, b;
    a.x = xrow[ka];
    a.y = xrow[ka + 1];
    b.x = W[(size_t)ka * NOUT + col0 + m];
    b.y = W[(size_t)(ka + 1) * NOUT + col0 + m];
    // 8 args: (neg_a, A, neg_b, B, c_mod, C, reuse_a, reuse_b)
    c = __builtin_amdgcn_wmma_f32_16x16x4_f32(false, a, false, b,
                                              (short)0, c, false, false);
  }

#pragma unroll
  for (int r = 0; r < 8; ++r) {
    const int mr  = r + 8 * half;
    const float s = dinv[row0 + mr];
    const float v = c[r] * s;
    const size_t idx = (size_t)(row0 + mr) * NOUT + col0 + m;
    hs[idx]  = v;   // pre-scaled features for the gather
    acc[idx] = v;   // self-loop contribution initializes the accumulator
  }
}

// ---------------------------------------------------------------------------
// Edge scatter: acc[dst] += ew * hs[src].
// One wave per edge; the 512B / 256B feature row is read with one b128/b64
// load per lane (perfectly coalesced, L2-resident: 25.6MB << 192MB L2),
// accumulation via native f32 atomics in L2.
// ---------------------------------------------------------------------------
__global__ void scatter128_kernel(const float* __restrict__ hs,
                                  const int* __restrict__ src,
                                  const int* __restrict__ dst,
                                  const float* __restrict__ ew,
                                  float* __restrict__ acc, int e) {
  const int lane = threadIdx.x & 31;
  const int edge = blockIdx.x * (blockDim.x >> 5) + (threadIdx.x >> 5);
  if (edge >= e) return;
  const int   s = src[edge];
  const int   d = dst[edge];
  const float w = ew[edge];
  const float4 v = ((const float4*)(hs + (size_t)s * 128))[lane];
  float* q = acc + (size_t)d * 128 + lane * 4;
  unsafeAtomicAdd(q + 0, w * v.x);
  unsafeAtomicAdd(q + 1, w * v.y);
  unsafeAtomicAdd(q + 2, w * v.z);
  unsafeAtomicAdd(q + 3, w * v.w);
}

__global__ void scatter64_kernel(const float* __restrict__ hs,
                                 const int* __restrict__ src,
                                 const int* __restrict__ dst,
                                 const float* __restrict__ ew,
                                 float* __restrict__ acc, int e) {
  const int lane = threadIdx.x & 31;
  const int edge = blockIdx.x * (blockDim.x >> 5) + (threadIdx.x >> 5);
  if (edge >= e) return;
  const int   s = src[edge];
  const int   d = dst[edge];
  const float w = ew[edge];
  const float2 v = ((const float2*)(hs + (size_t)s * 64))[lane];
  float* q = acc + (size_t)d * 64 + lane * 2;
  unsafeAtomicAdd(q + 0, w * v.x);
  unsafeAtomicAdd(q + 1, w * v.y);
}

// ---------------------------------------------------------------------------
// Finalize: out = (relu?) (dinv[row] * acc + bias[col])
// ---------------------------------------------------------------------------
template <int F>
__global__ void finalize_kernel(const float* __restrict__ acc,
                                const float* __restrict__ dinv,
                                const float* __restrict__ bias,
                                float* __restrict__ out, int n, int do_relu) {
  const size_t i = (size_t)blockIdx.x * blockDim.x + threadIdx.x;
  if (i >= (size_t)n * F) return;
  const int row = (int)(i / F);
  const int f   = (int)(i % F);
  float v = dinv[row] * acc[i] + bias[f];
  if (do_relu) v = fmaxf(v, 0.0f);
  out[i] = v;
}

// ---------------------------------------------------------------------------
// Launcher
// Inputs: z[N,64], edge_index[2,E] (int32), edge_attr[E], W1[64,128], b1[128],
//         W2[128,64], b2[64].  Output: [N,64] f32.
// Workspace layout (floats): deg[N] | dinv[N] | X1[N*128] | X2[N*128]
//   X1: hs1, later relu(h1).  X2: acc1, later split into hs2[N*64]|acc2[N*64].
// ---------------------------------------------------------------------------
extern "C" void kernel_launch(void* const* d_in, const int* in_sizes, int n_in,
                              void* d_out, int out_size, void* d_ws, size_t ws_size,
                              hipStream_t stream) {
  const float* z  = (const float*)d_in[0];
  const int*   ei = (const int*)d_in[1];
  const float* ew = (const float*)d_in[2];
  const float* W1 = (const float*)d_in[3];
  const float* b1 = (const float*)d_in[4];
  const float* W2 = (const float*)d_in[5];
  const float* b2 = (const float*)d_in[6];
  float* out = (float*)d_out;

  const int n = in_sizes[0] / 64;   // 50000
  const int e = in_sizes[2];        // 1600000
  const int* srcp = ei;             // edge_index[0]
  const int* dstp = ei + e;         // edge_index[1]

  float* ws   = (float*)d_ws;
  float* deg  = ws;
  float* dinv = deg + n;
  float* X1   = dinv + n;                 // N*128
  float* X2   = X1 + (size_t)n * 128;     // N*128
  float* hs2  = X2;                       // N*64 (reuses acc1 space)
  float* acc2 = X2 + (size_t)n * 64;      // N*64

  const int B = 256;  // 8 wave32 per block

  // 1) weighted in-degree (incl. self loop) and D^{-1/2}
  deg_init_kernel<<<(n + B - 1) / B, B, 0, stream>>>(deg, n);
  deg_accum_kernel<<<(e + B - 1) / B, B, 0, stream>>>(dstp, ew, deg, e);
  dinv_kernel<<<(n + B - 1) / B, B, 0, stream>>>(deg, dinv, n);

  // 2) layer 1 GEMM (WMMA f32) + row scaling; acc1 initialized to self-loop term
  {
    const long long waves = (long long)((n + 15) / 16) * (128 / 16);
    const long long thr   = waves * 32;
    gemm_wmma_scale_kernel<64, 128><<<(unsigned)((thr + B - 1) / B), B, 0, stream>>>(
        z, W1, dinv, X1, X2, n);
  }
  // 3) layer 1 edge aggregation
  {
    const long long thr = (long long)e * 32;
    scatter128_kernel<<<(unsigned)((thr + B - 1) / B), B, 0, stream>>>(
        X1, srcp, dstp, ew, X2, e);
  }
  // 4) layer 1 finalize: h1 = relu(dinv*acc1 + b1) -> X1
  {
    const long long thr = (long long)n * 128;
    finalize_kernel<128><<<(unsigned)((thr + B - 1) / B), B, 0, stream>>>(
        X2, dinv, b1, X1, n, 1);
  }
  // 5) layer 2 GEMM (WMMA f32) + row scaling
  {
    const long long waves = (long long)((n + 15) / 16) * (64 / 16);
    const long long thr   = waves * 32;
    gemm_wmma_scale_kernel<128, 64><<<(unsigned)((thr + B - 1) / B), B, 0, stream>>>(
        X1, W2, dinv, hs2, acc2, n);
  }
  // 6) layer 2 edge aggregation
  {
    const long long thr = (long long)e * 32;
    scatter64_kernel<<<(unsigned)((thr + B - 1) / B), B, 0, stream>>>(
        hs2, srcp, dstp, ew, acc2, e);
  }
  // 7) layer 2 finalize: out = dinv*acc2 + b2
  {
    const long long thr = (long long)n * 64;
    finalize_kernel<64><<<(unsigned)((thr + B - 1) / B), B, 0, stream>>>(
        acc2, dinv, b2, out, n, 0);
  }
}